// ODEFunc_51333449121987
// MI455X (gfx1250) — compile-verified
//
#include <hip/hip_runtime.h>
#include <stdint.h>

// Problem-shape assumption baked into the vector widths: D == 128
// (32 lanes x float4 == one feature row per wave32).
#define DIM 128

__device__ __forceinline__ float half_sigmoid(float a) {
  // 0.5 * sigmoid(a); v_rcp_f32 instead of the IEEE divide expansion
  return 0.5f * __builtin_amdgcn_rcpf(1.0f + __expf(-a));
}

// ---------------------------------------------------------------------------
// Pass 0: sig[i] = 0.5*sigmoid(alpha[i])  (into d_ws; 400KB, L2-resident)
// ---------------------------------------------------------------------------
__global__ void __launch_bounds__(256) sig_kernel(
    const float* __restrict__ alpha, float* __restrict__ sig, int n) {
  int i = blockIdx.x * 256 + threadIdx.x;
  if (i < n) sig[i] = half_sigmoid(alpha[i]);
}

// ---------------------------------------------------------------------------
// Pass 1: out[i,:] = x0[i,:] - 0.5*sigmoid(alpha[i]) * x[i,:]
// (so the edge pass can scatter s*val*x[col] straight into out)
// ---------------------------------------------------------------------------
__global__ void __launch_bounds__(256) init_kernel(
    const float* __restrict__ x, const float* __restrict__ x0,
    const float* __restrict__ alpha, float* __restrict__ out, int nvec) {
  int idx = blockIdx.x * 256 + threadIdx.x;   // one float4 per thread
  if (idx >= nvec) return;
  int row = idx >> 5;                          // D/4 == 32 float4 per row
  float s = half_sigmoid(alpha[row]);
  float4 xv = ((const float4*)x)[idx];
  float4 bv = ((const float4*)x0)[idx];
  float4 o;
  o.x = fmaf(-s, xv.x, bv.x);
  o.y = fmaf(-s, xv.y, bv.y);
  o.z = fmaf(-s, xv.z, bv.z);
  o.w = fmaf(-s, xv.w, bv.w);
  ((float4*)out)[idx] = o;
}

// ---------------------------------------------------------------------------
// Async-stage one 32-edge metadata tile (row[32] | col[32] | val[32], 384 B)
// into LDS with a single global_load_async_to_lds_b128 (lanes 0..23 carry
// 16 B each). Tracked by ASYNCcnt.
// ---------------------------------------------------------------------------
__device__ __forceinline__ void async_stage_tile(
    unsigned ldsByte, const int* rows, const int* cols, const float* vals,
    int tile, int lane) {
  int g = lane >> 3;         // 0: rows, 1: cols, 2: vals
  int lo = lane & 7;         // 8 lanes x 16 B = 128 B per array
  const char* src;
  if (g == 0)      src = (const char*)rows + ((size_t)tile << 7) + (lo << 4);
  else if (g == 1) src = (const char*)cols + ((size_t)tile << 7) + (lo << 4);
  else             src = (const char*)vals + ((size_t)tile << 7) + (lo << 4);
  unsigned dst = ldsByte + (unsigned)(g << 7) + (unsigned)(lo << 4);
  if (lane < 24) {
    asm volatile("global_load_async_to_lds_b128 %0, %1, off"
                 :: "v"(dst), "v"((unsigned long long)(uintptr_t)src)
                 : "memory");
  }
}

// ---------------------------------------------------------------------------
// Pass 2: for each edge e: out[row[e],:] += sig[row[e]]*val[e] * x[col[e],:]
// One wave per 32-edge tile; metadata double-buffered in LDS via async
// global->LDS copies; x-row gathers hit the 192MB L2 (x is only 51MB).
// USE_SIG: sig[] holds precomputed 0.5*sigmoid(alpha); else compute inline.
// ---------------------------------------------------------------------------
template <bool USE_SIG>
__global__ void __launch_bounds__(256) edge_kernel(
    const float* __restrict__ x, const float* __restrict__ alpha,
    const float* __restrict__ sig,
    const float* __restrict__ vals, const int* __restrict__ rows,
    const int* __restrict__ cols, float* __restrict__ out,
    int numTiles, int totalWaves) {
  __shared__ int smem[8 * 2 * 96];             // 8 waves x 2 bufs x 384 B
  const int lane  = threadIdx.x & 31;
  const int wib   = threadIdx.x >> 5;
  const int gwave = blockIdx.x * 8 + wib;

  int* wb0 = smem + wib * 192;
  // Generic->LDS address: low 32 bits of the flat address are the LDS offset.
  unsigned ldsByte0 = (unsigned)(uintptr_t)wb0;

  int t = gwave;
  if (t >= numTiles) return;
  int buf = 0;
  async_stage_tile(ldsByte0, rows, cols, vals, t, lane);

  const unsigned feat = (unsigned)(lane << 2);   // this lane's 4 features

  while (true) {
    int tn = t + totalWaves;
    bool more = tn < numTiles;
    if (more) {
      // prefetch next tile's metadata into the other buffer, then wait for
      // the current tile only (async loads complete in order)
      async_stage_tile(ldsByte0 + (unsigned)((buf ^ 1) * 384),
                       rows, cols, vals, tn, lane);
      asm volatile("s_wait_asynccnt 0x1" ::: "memory");
    } else {
      asm volatile("s_wait_asynccnt 0x0" ::: "memory");
    }

    const int* wb = wb0 + buf * 96;
#pragma unroll
    for (int jj = 0; jj < 8; ++jj) {           // 8 groups x 4 edges
      int4 r4 = *(const int4*)(wb + jj * 4);         // ds_load_b128
      int4 c4 = *(const int4*)(wb + 32 + jj * 4);
      int4 v4 = *(const int4*)(wb + 64 + jj * 4);
#pragma unroll
      for (int k = 0; k < 4; ++k) {
        // broadcast -> scalarize row/col for saddr+voffset addressing;
        // val only feeds VALU, leave it in a VGPR
        int row = __builtin_amdgcn_readfirstlane(((const int*)&r4)[k]);
        int col = __builtin_amdgcn_readfirstlane(((const int*)&c4)[k]);
        float v = __int_as_float(((const int*)&v4)[k]);
        float s = USE_SIG ? (v * sig[row])            // scalar load
                          : (v * half_sigmoid(alpha[row]));
        const float4 xv = *(const float4*)(x + (size_t)col * DIM + feat);
        float* o = out + (size_t)row * DIM + feat;
        __hip_atomic_fetch_add(o + 0, s * xv.x, __ATOMIC_RELAXED, __HIP_MEMORY_SCOPE_AGENT);
        __hip_atomic_fetch_add(o + 1, s * xv.y, __ATOMIC_RELAXED, __HIP_MEMORY_SCOPE_AGENT);
        __hip_atomic_fetch_add(o + 2, s * xv.z, __ATOMIC_RELAXED, __HIP_MEMORY_SCOPE_AGENT);
        __hip_atomic_fetch_add(o + 3, s * xv.w, __ATOMIC_RELAXED, __HIP_MEMORY_SCOPE_AGENT);
      }
    }
    if (!more) break;
    t = tn;
    buf ^= 1;
  }
}

// Tail (E % 32 edges): one block of 32 lanes per edge. Not launched for E=1.6M.
__global__ void edge_tail_kernel(
    const float* __restrict__ x, const float* __restrict__ alpha,
    const float* __restrict__ vals, const int* __restrict__ rows,
    const int* __restrict__ cols, float* __restrict__ out, int start) {
  int e = start + blockIdx.x;
  int lane = threadIdx.x;
  int row = rows[e], col = cols[e];
  float s = vals[e] * half_sigmoid(alpha[row]);
  const float4 xv = *(const float4*)(x + (size_t)col * DIM + (lane << 2));
  float* o = out + (size_t)row * DIM + (lane << 2);
  __hip_atomic_fetch_add(o + 0, s * xv.x, __ATOMIC_RELAXED, __HIP_MEMORY_SCOPE_AGENT);
  __hip_atomic_fetch_add(o + 1, s * xv.y, __ATOMIC_RELAXED, __HIP_MEMORY_SCOPE_AGENT);
  __hip_atomic_fetch_add(o + 2, s * xv.z, __ATOMIC_RELAXED, __HIP_MEMORY_SCOPE_AGENT);
  __hip_atomic_fetch_add(o + 3, s * xv.w, __ATOMIC_RELAXED, __HIP_MEMORY_SCOPE_AGENT);
}

extern "C" void kernel_launch(void* const* d_in, const int* in_sizes, int n_in,
                              void* d_out, int out_size, void* d_ws, size_t ws_size,
                              hipStream_t stream) {
  (void)n_in; (void)out_size;
  // inputs: 0=t(scalar), 1=x[N*D], 2=x0[N*D], 3=alpha[N], 4=vals[E],
  //         5=rows[E], 6=cols[E]
  const float* x     = (const float*)d_in[1];
  const float* x0    = (const float*)d_in[2];
  const float* alpha = (const float*)d_in[3];
  const float* vals  = (const float*)d_in[4];
  const int*   rows  = (const int*)d_in[5];
  const int*   cols  = (const int*)d_in[6];
  float* out = (float*)d_out;

  int ND = in_sizes[1];
  int N  = in_sizes[3];
  int E  = in_sizes[4];

  // Precompute 0.5*sigmoid(alpha) into workspace if it fits.
  bool useSig = (d_ws != nullptr) && (ws_size >= (size_t)N * sizeof(float));
  float* sigp = (float*)d_ws;
  if (useSig) {
    sig_kernel<<<(N + 255) / 256, 256, 0, stream>>>(alpha, sigp, N);
  }

  int nvec = ND >> 2;  // float4 count
  init_kernel<<<(nvec + 255) / 256, 256, 0, stream>>>(x, x0, alpha, out, nvec);

  int numTiles = E >> 5;
  if (numTiles > 0) {
    int blocks = (numTiles + 7) / 8;
    if (blocks > 2048) blocks = 2048;        // ~16K persistent waves
    if (useSig) {
      edge_kernel<true><<<blocks, 256, 0, stream>>>(
          x, alpha, sigp, vals, rows, cols, out, numTiles, blocks * 8);
    } else {
      edge_kernel<false><<<blocks, 256, 0, stream>>>(
          x, alpha, alpha, vals, rows, cols, out, numTiles, blocks * 8);
    }
  }
  int tail = numTiles << 5;
  if (tail < E) {
    edge_tail_kernel<<<E - tail, 32, 0, stream>>>(x, alpha, vals, rows, cols,
                                                  out, tail);
  }
}